// BayesianSparsePooler_20074677142320
// MI455X (gfx1250) — compile-verified
//
#include <hip/hip_runtime.h>

typedef __attribute__((ext_vector_type(2))) float v2f;
typedef __attribute__((ext_vector_type(4))) float v4f;
typedef __attribute__((ext_vector_type(8))) float v8f;

#define BATCH 8

// ---------------------------------------------------------------------------
// Kernel 1: out[b, r] = eps_b[r] * exp(b_log_var[r]) + b_mean[r]  for all b,
// implemented as a real WMMA rank-1 broadcast:
//   D(16x16) = A(16x4) * B(4x16),  A = ones in column K=0,  B row0 = bias tile
//   => D[m, n] = bias[tile + n] for every m;  rows m = 0..7 are the 8 batches.
// Also writes kl = 0 at out[BATCH*size2].
// ---------------------------------------------------------------------------
__global__ void bias_init_wmma(const float* __restrict__ b_mean,
                               const float* __restrict__ b_log_var,
                               const float* __restrict__ eps_b,
                               float* __restrict__ out,
                               int size2) {
  const int lane = threadIdx.x & 31;
  const int wave = threadIdx.x >> 5;
  const int tile = (blockIdx.x * (blockDim.x >> 5) + wave) << 4;  // 16 cols/wave
  const int n    = tile + (lane & 15);

  float bv = 0.0f;
  if (n < size2) bv = eps_b[n] * __expf(b_log_var[n]) + b_mean[n];

#if defined(__has_builtin) && __has_builtin(__builtin_amdgcn_wmma_f32_16x16x4_f32)
  // A layout (ISA 7.12.2, 32-bit A 16x4): lanes 0-15 -> {VGPR0:K=0, VGPR1:K=1},
  // lanes 16-31 -> {VGPR0:K=2, VGPR1:K=3}.  A[m,0] = 1, all other K = 0.
  v2f a;
  a.x = (lane < 16) ? 1.0f : 0.0f;
  a.y = 0.0f;
  // B (4x16): row K=0 carries the bias tile; VGPR1 = 0 kills the other rows.
  // (Rows that could alias into VGPR0's upper half are multiplied by A[m,k]=0.)
  v2f b;
  b.x = bv;
  b.y = 0.0f;
  v8f c = {};
  v8f d = __builtin_amdgcn_wmma_f32_16x16x4_f32(
      /*neg_a=*/false, a, /*neg_b=*/false, b,
      /*c_mod=*/(short)0, c, /*reuse_a=*/false, /*reuse_b=*/false);

  // C/D layout: lanes 0-15 hold N=lane, VGPR v holds row M=v (batches 0..7).
  if (lane < 16 && n < size2) {
#pragma unroll
    for (int v = 0; v < BATCH; ++v)
      out[(size_t)v * size2 + n] = d[v];
  }
#else
  // Scalar fallback (no wmma builtin available on this toolchain).
  if (lane < 16 && n < size2) {
#pragma unroll
    for (int v = 0; v < BATCH; ++v)
      out[(size_t)v * size2 + n] = bv;
  }
#endif

  if (blockIdx.x == 0 && threadIdx.x == 0)
    out[(size_t)BATCH * size2] = 0.0f;  // kl = 0
}

// ---------------------------------------------------------------------------
// Kernel 2: thread-per-edge scatter.  Memory-bound: 192 B of read-once weight
// streams + 8 B of indices per edge from HBM (all loaded non-temporal so the
// 192 MB L2 stays dedicated to the x-gather and atomic-output working sets),
// 8 L2-resident b128 gathers of x, and 32 global_atomic_add_f32 per edge into
// the bias-initialized output.
// ---------------------------------------------------------------------------
__global__ void edge_scatter(const float* __restrict__ x,
                             const float* __restrict__ w_mean,
                             const float* __restrict__ w_lv,
                             const float* __restrict__ eps_w,
                             const int*   __restrict__ rows,
                             const int*   __restrict__ cols,
                             float* __restrict__ out,
                             int nE, int size1, int size2) {
  const int e = blockIdx.x * blockDim.x + threadIdx.x;
  if (e >= nE) return;

  const long base = (long)e * 16;
  // Read-once index streams: non-temporal.
  const int rb = __builtin_nontemporal_load(rows + base);  // dst*4 + 0
  const int cb = __builtin_nontemporal_load(cols + base);  // src*4 + 0

  // values[e*16 + i*4 + j] = eps_w * exp(log_var) + mean
  // Read-once 96 MB weight streams: non-temporal b128 loads (native clang
  // vectors, since the builtin rejects HIP_vector_type structs).
  const v4f* m4 = (const v4f*)(w_mean + base);
  const v4f* l4 = (const v4f*)(w_lv   + base);
  const v4f* e4 = (const v4f*)(eps_w  + base);
  float v[16];
#pragma unroll
  for (int i = 0; i < 4; ++i) {
    v4f m  = __builtin_nontemporal_load(m4 + i);
    v4f l  = __builtin_nontemporal_load(l4 + i);
    v4f ew = __builtin_nontemporal_load(e4 + i);
    v[i * 4 + 0] = ew.x * __expf(l.x) + m.x;
    v[i * 4 + 1] = ew.y * __expf(l.y) + m.y;
    v[i * 4 + 2] = ew.z * __expf(l.z) + m.z;
    v[i * 4 + 3] = ew.w * __expf(l.w) + m.w;
  }

#pragma unroll
  for (int b = 0; b < BATCH; ++b) {
    // x is 2.56 MB total: default (RT) policy keeps it L2-resident.
    const v4f xv = *(const v4f*)(x + (size_t)b * size1 + cb);
    // s_j = sum_i v[i*4+j] * x[cb+i]
    const float s0 = v[0]*xv.x + v[4]*xv.y + v[ 8]*xv.z + v[12]*xv.w;
    const float s1 = v[1]*xv.x + v[5]*xv.y + v[ 9]*xv.z + v[13]*xv.w;
    const float s2 = v[2]*xv.x + v[6]*xv.y + v[10]*xv.z + v[14]*xv.w;
    const float s3 = v[3]*xv.x + v[7]*xv.y + v[11]*xv.z + v[15]*xv.w;
    float* o = out + (size_t)b * size2 + rb;
    atomicAdd(o + 0, s0);
    atomicAdd(o + 1, s1);
    atomicAdd(o + 2, s2);
    atomicAdd(o + 3, s3);
  }
}

// ---------------------------------------------------------------------------
extern "C" void kernel_launch(void* const* d_in, const int* in_sizes, int n_in,
                              void* d_out, int out_size, void* d_ws, size_t ws_size,
                              hipStream_t stream) {
  (void)n_in; (void)out_size; (void)d_ws; (void)ws_size;

  const float* x      = (const float*)d_in[0];
  const float* w_mean = (const float*)d_in[1];
  const float* w_lv   = (const float*)d_in[2];
  const float* b_mean = (const float*)d_in[3];
  const float* b_lv   = (const float*)d_in[4];
  const float* eps_w  = (const float*)d_in[5];
  const float* eps_b  = (const float*)d_in[6];
  const int*   rows   = (const int*)d_in[7];
  const int*   cols   = (const int*)d_in[8];
  float*       out    = (float*)d_out;

  const int size2 = in_sizes[3];            // 80000
  const int size1 = in_sizes[0] / BATCH;    // 80000
  const int nE    = in_sizes[1] / 16;       // 500000

  // 1) bias init (+ kl) — one 16-column tile per wave, 8 waves per block.
  const int tiles   = (size2 + 15) / 16;
  const int blocks1 = (tiles + 7) / 8;
  bias_init_wmma<<<blocks1, 256, 0, stream>>>(b_mean, b_lv, eps_b, out, size2);

  // 2) edge scatter-accumulate.
  const int blocks2 = (nE + 255) / 256;
  edge_scatter<<<blocks2, 256, 0, stream>>>(x, w_mean, w_lv, eps_w,
                                            rows, cols, out, nE, size1, size2);
}